// PDEModel_6133213299075
// MI455X (gfx1250) — compile-verified
//
#include <hip/hip_runtime.h>

#define NN   50000
#define EE   800000
#define HH   64
#define NBB  500
#define TT   10
#define STEPF 0.25f
#define SPU  4

typedef float v8f __attribute__((ext_vector_type(8)));
typedef float v2f __attribute__((ext_vector_type(2)));

// h0 = x * W1 (outer product N x 64); prime ACC=Y=h0, zero aggregation buffer.
__global__ void pde_init_kernel(const float* __restrict__ x, const float* __restrict__ W1,
                                float* __restrict__ Hb, float* __restrict__ Acc,
                                float* __restrict__ Y, float* __restrict__ G) {
    int i = blockIdx.x * blockDim.x + threadIdx.x;
    if (i >= NN * HH) return;
    int n = i >> 6, j = i & 63;
    float v = x[n] * W1[j];
    Hb[i] = v; Acc[i] = v; Y[i] = v; G[i] = 0.0f;
}

// G[dst] += w * Y[src] ; 16 lanes per edge (float4 per lane) -> coalesced gather.
__global__ void pde_scatter_kernel(const int* __restrict__ src, const int* __restrict__ dst,
                                   const float* __restrict__ w, const float* __restrict__ Y,
                                   float* __restrict__ G) {
    int idx = blockIdx.x * blockDim.x + threadIdx.x;
    int e = idx >> 4;
    if (e >= EE) return;
    int c = (idx & 15) << 2;
    int s = src[e];
    int d = dst[e];
    float ww = w[e];
    const float4 hv = *(const float4*)(Y + (size_t)s * HH + c);
    float* gp = G + (size_t)d * HH + c;
    atomicAdd(gp + 0, ww * hv.x);
    atomicAdd(gp + 1, ww * hv.y);
    atomicAdd(gp + 2, ww * hv.z);
    atomicAdd(gp + 3, ww * hv.w);
}

// K = G @ Wg - Y  via V_WMMA_F32_16X16X4_F32.
// Wave computes a 16x64 output tile: 16 K-chunks x 4 N-tiles = 64 WMMAs.
// A (16x4 f32): lanes 0-15 row M=lane, K={0,1}; lanes 16-31 row M=lane-16, K={2,3}.
// B (4x16 f32): lanes 0-15 col N=lane, K={0,1}; lanes 16-31 col N=lane-16, K={2,3}.
// C/D (16x16 f32): VGPR r holds M=r (lanes 0-15) / M=r+8 (lanes 16-31), N=lane&15.
__global__ void __launch_bounds__(256)
pde_gemm_kernel(const float* __restrict__ G, const float* __restrict__ Y,
                const float* __restrict__ Wg, float* __restrict__ K) {
    __shared__ float sW[HH * HH];
    {
        const float4* s = (const float4*)Wg;
        float4* d = (float4*)sW;
        for (int i = threadIdx.x; i < (HH * HH) / 4; i += 256) d[i] = s[i];
    }
    __syncthreads();

    int wave = threadIdx.x >> 5;
    int lane = threadIdx.x & 31;
    int rowBase = blockIdx.x * 128 + wave * 16;
    if (rowBase >= NN) return;           // wave-uniform; NN % 16 == 0 so live waves are full tiles

    int lh  = lane & 15;                 // row (A) / col (B,D) within tile
    int khi = (lane >> 4) << 1;          // 0 or 2: which K-pair this half-wave owns

    v8f c0 = {}, c1 = {}, c2 = {}, c3 = {};
    const float* Arow = G + (size_t)(rowBase + lh) * HH + khi;

    #pragma unroll
    for (int k0 = 0; k0 < HH; k0 += 4) {
        v2f a;
        a.x = Arow[k0];
        a.y = Arow[k0 + 1];
        const float* bp = sW + (k0 + khi) * HH + lh;
        v2f b0, b1, b2, b3;
        b0.x = bp[0];  b0.y = bp[HH + 0];
        b1.x = bp[16]; b1.y = bp[HH + 16];
        b2.x = bp[32]; b2.y = bp[HH + 32];
        b3.x = bp[48]; b3.y = bp[HH + 48];
        c0 = __builtin_amdgcn_wmma_f32_16x16x4_f32(false, a, false, b0, (short)0, c0, false, false);
        c1 = __builtin_amdgcn_wmma_f32_16x16x4_f32(false, a, false, b1, (short)0, c1, false, false);
        c2 = __builtin_amdgcn_wmma_f32_16x16x4_f32(false, a, false, b2, (short)0, c2, false, false);
        c3 = __builtin_amdgcn_wmma_f32_16x16x4_f32(false, a, false, b3, (short)0, c3, false, false);
    }

    int mhi = (lane >> 4) << 3;          // 0 or 8
    #pragma unroll
    for (int r = 0; r < 8; ++r) {
        int row = rowBase + r + mhi;
        float* outp = K + (size_t)row * HH;
        const float* yp = Y + (size_t)row * HH;
        outp[lh]      = c0[r] - yp[lh];
        outp[lh + 16] = c1[r] - yp[lh + 16];
        outp[lh + 32] = c2[r] - yp[lh + 32];
        outp[lh + 48] = c3[r] - yp[lh + 48];
    }
}

// K[boundary] = bc - Y[boundary]
__global__ void pde_boundary_kernel(const int* __restrict__ bidx, const float* __restrict__ bc,
                                    const float* __restrict__ Y, float* __restrict__ K) {
    int i = blockIdx.x * blockDim.x + threadIdx.x;
    if (i >= NBB * HH) return;
    int b = i >> 6, j = i & 63;
    int node = bidx[b];
    K[(size_t)node * HH + j] = bc[b] - Y[(size_t)node * HH + j];
}

// RK4 stage combine: ACC += wacc*K; Y = H + cnext*K (or H=Y=ACC on final stage); G = 0.
__global__ void pde_combine_kernel(float* __restrict__ Hb, float* __restrict__ Acc,
                                   float* __restrict__ Y, float* __restrict__ G,
                                   const float* __restrict__ K,
                                   float wacc, float cnext, int final_stage) {
    int i = blockIdx.x * blockDim.x + threadIdx.x;
    if (i >= NN * HH) return;
    float kv = K[i];
    float a = Acc[i] + wacc * kv;
    Acc[i] = a;
    if (final_stage) { Hb[i] = a; Y[i] = a; }
    else             { Y[i] = Hb[i] + cnext * kv; }
    G[i] = 0.0f;
}

// out[n*(T+1)+t] = relu(h[n,:]) . W2
__global__ void pde_snapshot_kernel(const float* __restrict__ Hb, const float* __restrict__ W2,
                                    float* __restrict__ out, int t) {
    int n = blockIdx.x * blockDim.x + threadIdx.x;
    if (n >= NN) return;
    const float4* hp = (const float4*)(Hb + (size_t)n * HH);
    const float4* wp = (const float4*)W2;
    float acc = 0.0f;
    #pragma unroll
    for (int q = 0; q < HH / 4; ++q) {
        float4 h = hp[q];
        float4 w = wp[q];
        acc += fmaxf(h.x, 0.0f) * w.x + fmaxf(h.y, 0.0f) * w.y
             + fmaxf(h.z, 0.0f) * w.z + fmaxf(h.w, 0.0f) * w.w;
    }
    out[(size_t)n * (TT + 1) + t] = acc;
}

extern "C" void kernel_launch(void* const* d_in, const int* in_sizes, int n_in,
                              void* d_out, int out_size, void* d_ws, size_t ws_size,
                              hipStream_t stream) {
    const float* x   = (const float*)d_in[0];
    const int*   eix = (const int*)  d_in[1];   // (2, E): src then dst
    const float* ew  = (const float*)d_in[2];
    const float* bc  = (const float*)d_in[3];
    const int*   bix = (const int*)  d_in[4];
    const float* W1  = (const float*)d_in[5];
    const float* Wg  = (const float*)d_in[6];
    const float* W2  = (const float*)d_in[7];
    float* out = (float*)d_out;

    const int* src = eix;
    const int* dst = eix + EE;

    size_t nh = (size_t)NN * HH;
    float* Hb  = (float*)d_ws;        // current state h
    float* Acc = Hb  + nh;            // RK4 accumulator
    float* Y   = Acc + nh;            // evaluation point
    float* G   = Y   + nh;            // scatter aggregation
    float* K   = G   + nh;            // stage derivative

    dim3 b256(256);
    const int gNH   = (NN * HH + 255) / 256;
    const int gN    = (NN + 255) / 256;
    const int gScat = (EE * 16 + 255) / 256;
    const int gGemm = (NN + 127) / 128;
    const int gB    = (NBB * HH + 255) / 256;

    const float wa[4] = { STEPF / 6.0f, STEPF / 3.0f, STEPF / 3.0f, STEPF / 6.0f };
    const float cn[4] = { 0.5f * STEPF, 0.5f * STEPF, STEPF, 0.0f };

    pde_init_kernel<<<gNH, b256, 0, stream>>>(x, W1, Hb, Acc, Y, G);
    pde_snapshot_kernel<<<gN, b256, 0, stream>>>(Hb, W2, out, 0);

    for (int t = 1; t <= TT; ++t) {
        for (int s = 0; s < SPU; ++s) {
            for (int st = 0; st < 4; ++st) {
                pde_scatter_kernel<<<gScat, b256, 0, stream>>>(src, dst, ew, Y, G);
                pde_gemm_kernel<<<gGemm, b256, 0, stream>>>(G, Y, Wg, K);
                pde_boundary_kernel<<<gB, b256, 0, stream>>>(bix, bc, Y, K);
                pde_combine_kernel<<<gNH, b256, 0, stream>>>(Hb, Acc, Y, G, K,
                                                             wa[st], cn[st], st == 3);
            }
        }
        pde_snapshot_kernel<<<gN, b256, 0, stream>>>(Hb, W2, out, t);
    }
}